// SmartEnsembleModel_66005057405199
// MI455X (gfx1250) — compile-verified
//
#include <hip/hip_runtime.h>
#include <stdint.h>

// Problem constants (B=16, H=W=512)
#define IMG_H 512
#define IMG_W 512
#define IMG_HW (IMG_H * IMG_W)   // 262144 = 2^18
#define NBATCH 16
#define NPIX (NBATCH * IMG_HW)   // 4194304
#define TILE 64
#define HALO 4
#define TDIM (TILE + 2 * HALO)   // 72
#define ROW_DW (TDIM / 4)        // 18 dwords per tile row
#define MIN_SIZE 262             // keep components with size > 262

typedef unsigned char u8;
typedef unsigned int u32;

// ---------------------------------------------------------------------------
// Raw LDS byte offset of a __shared__ object (addrspace(3) ptrtoint), used as
// the VDST operand of global_load_async_to_lds_* (LDS address VGPR).
// ---------------------------------------------------------------------------
__device__ __forceinline__ u32 lds_off_of(const void* p) {
  typedef __attribute__((address_space(3))) const u8 lds_u8;
  return (u32)(uintptr_t)(lds_u8*)p;
}

// ---------------------------------------------------------------------------
// Kernel 1: majority vote (fallback to pred1 on 1-vote disagreement) fused
// with union-find state init (parent[g]=g, cnt[g]=0) to save a full pass.
// ---------------------------------------------------------------------------
__global__ __launch_bounds__(256) void vote_init_kernel(
    const float* __restrict__ x, u8* __restrict__ m0, int* __restrict__ parent,
    int* __restrict__ cnt) {
  int p = blockIdx.x * blockDim.x + threadIdx.x;
  if (p >= NPIX) return;
  int b = p >> 18;            // / IMG_HW
  int r = p & (IMG_HW - 1);   // % IMG_HW
  const float* xb = x + (size_t)b * 4 * IMG_HW;
  int p1 = xb[1 * IMG_HW + r] > 0.5f;
  int p2 = xb[2 * IMG_HW + r] > 0.5f;
  int p3 = xb[3 * IMG_HW + r] > 0.5f;
  int v = p1 + p2 + p3;
  m0[p] = (u8)((v >= 2) ? 1 : ((v == 1) ? p1 : 0));
  parent[p] = p;  // re-initialized every call: deterministic across replays
  cnt[p] = 0;
}

// ---------------------------------------------------------------------------
// Kernel 2: fused morphology close-then-open = dilate, erode, erode, dilate
// (3x3 cross). One 64x64 tile per workgroup with a 4-px halo staged into LDS
// via CDNA5 async global->LDS copies. Border padding is handled logically
// (dilate pads 0, erode pads 1) from global coordinates; LDS cells that were
// never loaded (outside the image) are never read.
// ---------------------------------------------------------------------------
template <bool CHECKED>
__device__ __forceinline__ int mval(const u8* buf, int li, int lj, int gi0,
                                    int gj0, int pad) {
  if (CHECKED) {
    int gi = gi0 + li, gj = gj0 + lj;
    if (gi < 0 || gi >= IMG_H || gj < 0 || gj >= IMG_W) return pad;
  }
  return (int)buf[li * TDIM + lj];
}

template <bool CHECKED>
__device__ __forceinline__ u8* morph_steps(u8* bufA, u8* bufB, int gi0,
                                           int gj0, int tid, int nth) {
  u8* cur = bufA;
  u8* nxt = bufB;
#pragma unroll
  for (int s = 0; s < 4; ++s) {
    const int lo = s + 1;
    const int n = TDIM - 2 * (s + 1);          // 70, 68, 66, 64
    const int pad = (s == 1 || s == 2) ? 1 : 0;  // dilate,erode,erode,dilate
    for (int t = tid; t < n * n; t += nth) {
      const int q = t / n;
      const int li = lo + q;
      const int lj = lo + (t - q * n);
      const int c  = mval<CHECKED>(cur, li,     lj,     gi0, gj0, pad);
      const int up = mval<CHECKED>(cur, li - 1, lj,     gi0, gj0, pad);
      const int dn = mval<CHECKED>(cur, li + 1, lj,     gi0, gj0, pad);
      const int lf = mval<CHECKED>(cur, li,     lj - 1, gi0, gj0, pad);
      const int rt = mval<CHECKED>(cur, li,     lj + 1, gi0, gj0, pad);
      // binary values: OR == max (dilate), AND == min (erode)
      const int rv = pad ? (c & up & dn & lf & rt) : (c | up | dn | lf | rt);
      nxt[li * TDIM + lj] = (u8)rv;
    }
    __syncthreads();
    u8* tmp = cur; cur = nxt; nxt = tmp;
  }
  return cur;
}

__global__ __launch_bounds__(256) void morph_kernel(const u8* __restrict__ m0,
                                                    u8* __restrict__ mout) {
  __shared__ u8 bufA[TDIM * TDIM];
  __shared__ u8 bufB[TDIM * TDIM];
  const int b = blockIdx.z;
  const int gi0 = (int)blockIdx.y * TILE - HALO;
  const int gj0 = (int)blockIdx.x * TILE - HALO;
  const int tid = threadIdx.x;
  const int nth = blockDim.x;

  // ---- async tile load: 72 rows x 18 dwords, EXEC-predicated per dword.
  // gj0 = 64k-4 is dword-aligned and W is a multiple of 4, so every dword is
  // entirely in-image or entirely out-of-image. Each wave issues ~6 async
  // copies (ASYNCcnt well under 63), waits its own counter, then barriers.
  const unsigned long long base = (unsigned long long)m0;
  const u32 ldsA = lds_off_of(bufA);
  for (int t = tid; t < TDIM * ROW_DW; t += nth) {
    const int li = t / ROW_DW;
    const int dj = t - li * ROW_DW;
    const int gi = gi0 + li;
    const int gj = gj0 + dj * 4;
    if (gi >= 0 && gi < IMG_H && gj >= 0 && gj < IMG_W) {
      const u32 goff = (u32)(b * IMG_HW + gi * IMG_W + gj);
      const u32 loff = ldsA + (u32)(li * TDIM + dj * 4);
      asm volatile("global_load_async_to_lds_b32 %0, %1, %2"
                   :
                   : "v"(loff), "v"(goff), "s"(base)
                   : "memory");
    }
  }
  asm volatile("s_wait_asynccnt 0" ::: "memory");
  __syncthreads();

  // ---- 4 cross ops in LDS; interior tiles skip all border clamping ----
  const bool interior = (gi0 >= 0) && (gj0 >= 0) && (gi0 + TDIM <= IMG_H) &&
                        (gj0 + TDIM <= IMG_W);
  u8* cur = interior ? morph_steps<false>(bufA, bufB, gi0, gj0, tid, nth)
                     : morph_steps<true>(bufA, bufB, gi0, gj0, tid, nth);

  // ---- write interior 64x64 ----
  for (int t = tid; t < TILE * TILE; t += nth) {
    const int li = HALO + (t >> 6);
    const int lj = HALO + (t & (TILE - 1));
    const int gi = gi0 + li;
    const int gj = gj0 + lj;
    mout[b * IMG_HW + gi * IMG_W + gj] = cur[li * TDIM + lj];
  }
}

// ---------------------------------------------------------------------------
// Lock-free union-find (min-root policy -> deterministic final labeling).
// ---------------------------------------------------------------------------
__device__ __forceinline__ int uf_find(int* P, int i) {
  int p = P[i];
  while (p != i) {
    int gp = P[p];
    if (gp == p) return p;
    P[i] = gp;  // path halving (benign race)
    i = gp;
    p = P[i];
  }
  return i;
}

__device__ void uf_union(int* P, int a, int b) {
  a = uf_find(P, a);
  b = uf_find(P, b);
  while (a != b) {
    if (a < b) { int t = a; a = b; b = t; }  // larger index points to smaller
    int old = atomicCAS(&P[a], a, b);
    if (old == a) return;
    a = uf_find(P, old);
    b = uf_find(P, b);
  }
}

__global__ __launch_bounds__(256) void ccl_link_kernel(const u8* __restrict__ m,
                                                       int* __restrict__ parent) {
  int g = blockIdx.x * blockDim.x + threadIdx.x;
  if (g >= NPIX) return;
  if (!m[g]) return;
  int r = g & (IMG_HW - 1);
  int j = r & (IMG_W - 1);
  int i = r >> 9;  // / IMG_W
  if (j + 1 < IMG_W && m[g + 1]) uf_union(parent, g, g + 1);
  if (i + 1 < IMG_H && m[g + IMG_W]) uf_union(parent, g, g + IMG_W);
}

__global__ __launch_bounds__(256) void ccl_count_kernel(const u8* __restrict__ m,
                                                        int* __restrict__ parent,
                                                        int* __restrict__ cnt) {
  int g = blockIdx.x * blockDim.x + threadIdx.x;
  if (g >= NPIX) return;
  if (!m[g]) return;
  atomicAdd(&cnt[uf_find(parent, g)], 1);
}

__global__ __launch_bounds__(256) void output_kernel(const u8* __restrict__ m,
                                                     int* __restrict__ parent,
                                                     const int* __restrict__ cnt,
                                                     float* __restrict__ out) {
  int g = blockIdx.x * blockDim.x + threadIdx.x;
  if (g >= NPIX) return;
  float v = 0.0f;
  if (m[g]) {
    int root = uf_find(parent, g);
    if (cnt[root] > MIN_SIZE) v = 1.0f;
  }
  out[g] = v;  // write every element: d_out is poisoned before timing
}

// ---------------------------------------------------------------------------
// Launcher. Workspace layout (40 MB total):
//   [0,16MB)  parent (int32 x NPIX)
//   [16,32MB) cnt    (int32 x NPIX)
//   [32,36MB) m0     (u8 vote mask)
//   [36,40MB) m1     (u8 final morphology mask)
// ---------------------------------------------------------------------------
extern "C" void kernel_launch(void* const* d_in, const int* in_sizes, int n_in,
                              void* d_out, int out_size, void* d_ws,
                              size_t ws_size, hipStream_t stream) {
  (void)in_sizes; (void)n_in; (void)out_size; (void)ws_size;
  const float* x = (const float*)d_in[0];
  float* out = (float*)d_out;

  int* parent = (int*)d_ws;
  int* cnt = parent + NPIX;
  u8* m0 = (u8*)(cnt + NPIX);
  u8* m1 = m0 + NPIX;

  const int threads = 256;
  const int blocks = (NPIX + threads - 1) / threads;  // 16384

  vote_init_kernel<<<blocks, threads, 0, stream>>>(x, m0, parent, cnt);
  morph_kernel<<<dim3(IMG_W / TILE, IMG_H / TILE, NBATCH), threads, 0, stream>>>(
      m0, m1);
  ccl_link_kernel<<<blocks, threads, 0, stream>>>(m1, parent);
  ccl_count_kernel<<<blocks, threads, 0, stream>>>(m1, parent, cnt);
  output_kernel<<<blocks, threads, 0, stream>>>(m1, parent, cnt, out);
}